// SelfAttention_74646531605092
// MI455X (gfx1250) — compile-verified
//
#include <hip/hip_runtime.h>
#include <hip/hip_bf16.h>
#include <math.h>

// out = softmax_s( x @ Wq_x ),  Wq_x[k,h] = sum_d W[k, h*D+d] * q[h,d]
// (DeepSets MLP / set-feature / bias terms are constant over the softmax axis
//  and cancel exactly.)

typedef __attribute__((ext_vector_type(2))) float v2f;
typedef __attribute__((ext_vector_type(8))) float v8f;

#define SS 512   // set size
#define DD 128   // in_size
#define HH 8     // heads
#define NP 16    // padded head dim for 16x16 WMMA tiles
#define GROUPS 256  // B*N

// ---------------------------------------------------------------------------
// Kernel 1: Wq[k][n] = sum_d W[k, n*128+d] * q[n, d]   (k<128, n<8; pad n..15=0)
// ---------------------------------------------------------------------------
__global__ __launch_bounds__(256) void wq_project_kernel(
    const float* __restrict__ W, const float* __restrict__ q,
    float* __restrict__ Wq)
{
  int t = blockIdx.x * blockDim.x + threadIdx.x;   // 2048 slots = 128*16
  if (t >= DD * NP) return;
  int k = t >> 4;
  int n = t & 15;
  float acc = 0.0f;
  if (n < HH) {
    const float* wrow = W + (size_t)k * (HH * DD) + n * DD;
    const float* qrow = q + n * DD;
#pragma unroll 8
    for (int d = 0; d < DD; ++d) acc = fmaf(wrow[d], qrow[d], acc);
  }
  Wq[t] = acc;   // K-major layout [k][n], zero-padded columns 8..15
}

// ---------------------------------------------------------------------------
// Kernel 2: per (b,n) group: E = x_g(512x128) @ Wq(128x16) via
// V_WMMA_F32_16X16X4_F32, then softmax over the 512 set rows per head.
// Block = 256 threads (8 waves); each wave owns 4 row-tiles of 16.
// ---------------------------------------------------------------------------
__global__ __launch_bounds__(256) void attn_softmax_kernel(
    const float* __restrict__ x, const float* __restrict__ Wq,
    float* __restrict__ out)
{
  __shared__ float sWq[DD * NP];    // 8 KB staged padded Wq
  __shared__ float sPart[8][NP];    // per-wave partials
  __shared__ float sMax[NP];
  __shared__ float sSum[NP];

  const int tid  = threadIdx.x;
  const int wave = tid >> 5;
  const int lane = tid & 31;
  const int col  = lane & 15;   // N column (head id for col<8)
  const int hi   = lane >> 4;   // lane half selects K/M sub-slice
  const int ksub = hi * 2;

  for (int i = tid; i < DD * NP; i += 256) sWq[i] = Wq[i];
  __syncthreads();

  const float* xg = x + (size_t)blockIdx.x * (SS * DD);

  // ---- GEMM: acc[t] holds a 16x16 f32 C/D tile (rows m0..m0+15, cols 0..15)
  v8f acc[4];
#pragma unroll
  for (int t = 0; t < 4; ++t) {
    const int m0 = wave * 64 + t * 16;
    // A 16x4 f32 layout: lanes 0-15 -> M=lane, K=0,1 ; lanes 16-31 -> M=lane-16, K=2,3
    const float* arow = xg + (size_t)(m0 + col) * DD + ksub;
    if (t < 3) __builtin_prefetch(arow + 16 * DD, 0, 0);  // next row-tile
    v8f c = {};
#pragma unroll
    for (int k0 = 0; k0 < DD; k0 += 4) {
      v2f a = *(const v2f*)(arow + k0);          // x[m0+col][k0+ksub .. +1]
      v2f b;                                      // B 4x16: K rows ksub,ksub+1
      b.x = sWq[(k0 + ksub    ) * NP + col];
      b.y = sWq[(k0 + ksub + 1) * NP + col];
      c = __builtin_amdgcn_wmma_f32_16x16x4_f32(
              /*neg_a=*/false, a, /*neg_b=*/false, b,
              /*c_mod=*/(short)0, c, /*reuse_a=*/false, /*reuse_b=*/false);
    }
    acc[t] = c;
  }

  // ---- softmax over s (512 rows spread across 8 waves x 2 lane-halves) ----
  float lmax = -3.402823466e38f;
#pragma unroll
  for (int t = 0; t < 4; ++t)
#pragma unroll
    for (int j = 0; j < 8; ++j) lmax = fmaxf(lmax, acc[t][j]);
  lmax = fmaxf(lmax, __shfl_xor(lmax, 16, 32));   // merge the two halves
  if (lane < 16) sPart[wave][col] = lmax;
  __syncthreads();
  if (tid < NP) {
    float m = sPart[0][tid];
#pragma unroll
    for (int w = 1; w < 8; ++w) m = fmaxf(m, sPart[w][tid]);
    sMax[tid] = m;
  }
  __syncthreads();

  const float mval = sMax[col];
  float lsum = 0.0f;
#pragma unroll
  for (int t = 0; t < 4; ++t)
#pragma unroll
    for (int j = 0; j < 8; ++j) {
      float e = __expf(acc[t][j] - mval);
      acc[t][j] = e;
      lsum += e;
    }
  lsum += __shfl_xor(lsum, 16, 32);
  if (lane < 16) sPart[wave][col] = lsum;
  __syncthreads();
  if (tid < NP) {
    float s = sPart[0][tid];
#pragma unroll
    for (int w = 1; w < 8; ++w) s += sPart[w][tid];
    sSum[tid] = s;
  }
  __syncthreads();

  // ---- normalize + store: out[group, s, h], h fastest ----
  if (col < HH) {
    const float inv = 1.0f / sSum[col];
    float* og = out + (size_t)blockIdx.x * (SS * HH);
#pragma unroll
    for (int t = 0; t < 4; ++t) {
      const int m0 = wave * 64 + t * 16;
#pragma unroll
      for (int j = 0; j < 8; ++j) {
        const int s = m0 + j + 8 * hi;   // C/D layout: vgpr j -> row j (+8 for hi half)
        og[s * HH + col] = acc[t][j] * inv;
      }
    }
  }
}

// ---------------------------------------------------------------------------
extern "C" void kernel_launch(void* const* d_in, const int* in_sizes, int n_in,
                              void* d_out, int out_size, void* d_ws, size_t ws_size,
                              hipStream_t stream) {
  // setup_inputs order: x, w1, b1, w2, b2, w3, b3, W, Wb, q
  const float* x = (const float*)d_in[0];
  const float* W = (const float*)d_in[7];
  const float* q = (const float*)d_in[9];
  float* Wq  = (float*)d_ws;       // 128*16 floats = 8 KB scratch
  float* out = (float*)d_out;      // (B,N,S,H) = 1,048,576 floats

  wq_project_kernel<<<8, 256, 0, stream>>>(W, q, Wq);
  attn_softmax_kernel<<<GROUPS, 256, 0, stream>>>(x, Wq, out);
}